// NucleusSampling_79336635892529
// MI455X (gfx1250) — compile-verified
//
#include <hip/hip_runtime.h>
#include <stdint.h>

// Nucleus (top-p) sampling, sort-free, one workgroup per row.
// B=256 rows, V=128000 fp32 logits. Multi-pass histogram threshold search;
// passes 2..4 hit the 192MB L2 (whole input = 131MB resident).
//
// CDNA5-specific paths used:
//  - global_load_async_to_lds_b128 + s_wait_asynccnt/s_wait_dscnt (ASYNCcnt engine)
//  - ds_add_u64 LDS atomics (deterministic fixed-point histogram)
//  - wave32 shuffles / ballot

#define NTHREADS 512
#define NWAVES   16
#define TILE     2048            // floats per async tile (8 KB), 16 B per lane
#define NBINS    4096
#define SUBBINS  1024
#define BINSCALE 128.0f          // NBINS / 32.0
#define WBIN     0.0078125f      // 32.0 / NBINS
#define SUBSCALE 131072.0f       // SUBBINS / WBIN
#define TOPP     0.9
#define FIXSCALE 1.099511627776e12f   // 2^40  (Q40 fixed point, sum < 2^52 here)
#define NEGINF   -3.0e38f

__device__ __forceinline__ float waveMaxF(float v) {
#pragma unroll
  for (int o = 16; o > 0; o >>= 1) v = fmaxf(v, __shfl_xor(v, o, 32));
  return v;
}
__device__ __forceinline__ float waveSumF(float v) {
#pragma unroll
  for (int o = 16; o > 0; o >>= 1) v += __shfl_xor(v, o, 32);
  return v;
}
__device__ __forceinline__ float blockMaxF(float v, float* red) {
  const int lane = threadIdx.x & 31, wid = threadIdx.x >> 5;
  v = waveMaxF(v);
  if (lane == 0) red[wid] = v;
  __syncthreads();
  if (wid == 0) {
    float r = (lane < NWAVES) ? red[lane] : NEGINF;
    r = waveMaxF(r);
    if (lane == 0) red[0] = r;
  }
  __syncthreads();
  float out = red[0];
  __syncthreads();
  return out;
}

struct Sel { float m, invT, cbLow; int cb, sj, all; };

// Weight of token in the renormalized (masked) softmax; 0 if not selected.
__device__ __forceinline__ float selW(float logit, const Sel& S) {
  float d = fmaxf(S.m - logit * S.invT, 0.0f);
  float w = __expf(-d);
  if (S.all) return w;
  int bin = (int)(d * BINSCALE); bin = bin < (NBINS - 1) ? bin : (NBINS - 1);
  if (bin < S.cb) return w;
  if (bin > S.cb) return 0.0f;
  int sub = (int)((d - S.cbLow) * SUBSCALE);
  sub = sub < 0 ? 0 : (sub > (SUBBINS - 1) ? (SUBBINS - 1) : sub);
  return (sub <= S.sj) ? w : 0.0f;
}

__global__ __launch_bounds__(NTHREADS, 1)
void NucleusSampling_79336635892529_kernel(const float* __restrict__ logits,
                                           const float* __restrict__ uvec,
                                           const int* __restrict__ temp,
                                           int* __restrict__ out, int V) {
  __shared__ float stage[2][TILE];                 // async-copy staging (16 KB)
  __shared__ unsigned long long hist[NBINS];       // Q40 histogram (32 KB)
  __shared__ float chunkSum[256];
  __shared__ float red[NWAVES];
  __shared__ int   shCb, shSj, shAll, shCross, shToken;
  __shared__ unsigned long long shRbin;
  __shared__ float shRem;

  const int tid = threadIdx.x, lane = tid & 31, wid = tid >> 5;
  const int b = blockIdx.x;
  const float* __restrict__ row = logits + (size_t)b * (size_t)V;
  const float invT = 1.0f / (float)temp[0];

  // ---------------- Phase A: row max (HBM stream via async global->LDS) ----
  // Each lane async-copies its own 16B into LDS and reads back only its own
  // bytes, so only per-wave counter waits are needed (no barriers).
  float mloc = NEGINF;
  const int nFull = V / TILE;
  const uint32_t lds0 = (uint32_t)(uintptr_t)&stage[0][tid * 4];
  const uint32_t lds1 = (uint32_t)(uintptr_t)&stage[1][tid * 4];
  if (nFull > 0) {
    asm volatile("global_load_async_to_lds_b128 %0, %1, off"
                 :: "v"(lds0), "v"((uint64_t)(uintptr_t)(row + tid * 4))
                 : "memory");
  }
  for (int t = 0; t < nFull; ++t) {
    if (t + 1 < nFull) {
      // make sure prior LDS reads of the buffer we are about to overwrite
      // have completed before the async engine can land new data there
      asm volatile("s_wait_dscnt 0" ::: "memory");
      const uint64_t ga =
          (uint64_t)(uintptr_t)(row + (size_t)(t + 1) * TILE + tid * 4);
      const uint32_t ldst = (t & 1) ? lds0 : lds1;   // buffer (t+1)&1
      asm volatile("global_load_async_to_lds_b128 %0, %1, off"
                   :: "v"(ldst), "v"(ga) : "memory");
      asm volatile("s_wait_asynccnt 1" ::: "memory"); // oldest (tile t) done
    } else {
      asm volatile("s_wait_asynccnt 0" ::: "memory");
    }
    const float4 v = *(const float4*)&stage[t & 1][tid * 4];
    mloc = fmaxf(mloc, fmaxf(fmaxf(v.x, v.y), fmaxf(v.z, v.w)));
  }
  for (int i = nFull * TILE + tid; i < V; i += NTHREADS)
    mloc = fmaxf(mloc, row[i]);
  mloc *= invT;                       // invT > 0: max commutes with scaling
  const float m = blockMaxF(mloc, red);

  // ---------------- Phase B: Q40 histogram of prob mass vs (m - l/T) -------
  for (int i = tid; i < NBINS; i += NTHREADS) hist[i] = 0ull;
  __syncthreads();
  const int nV4 = V >> 2;
  const float4* __restrict__ row4 = (const float4*)row;
  for (int i = tid; i < nV4; i += NTHREADS) {
    float4 v = row4[i];
#pragma unroll
    for (int k = 0; k < 4; ++k) {
      float d = fmaxf(m - ((const float*)&v)[k] * invT, 0.0f);
      int bin = (int)(d * BINSCALE); bin = bin < (NBINS - 1) ? bin : (NBINS - 1);
      unsigned long long wf = (unsigned long long)(__expf(-d) * FIXSCALE);
      if (wf) atomicAdd(&hist[bin], wf);            // ds_add_u64
    }
  }
  for (int i = (nV4 << 2) + tid; i < V; i += NTHREADS) {
    float d = fmaxf(m - row[i] * invT, 0.0f);
    int bin = (int)(d * BINSCALE); bin = bin < (NBINS - 1) ? bin : (NBINS - 1);
    unsigned long long wf = (unsigned long long)(__expf(-d) * FIXSCALE);
    if (wf) atomicAdd(&hist[bin], wf);
  }
  __syncthreads();
  if (tid == 0) {
    unsigned long long Z = 0;
    for (int i = 0; i < NBINS; ++i) Z += hist[i];   // Z from same fixed-point sums
    unsigned long long tZ = (unsigned long long)(TOPP * (double)Z);
    unsigned long long cum = 0; int cb = -1; unsigned long long rb = 0;
    for (int i = 0; i < NBINS; ++i) {
      if (cum + hist[i] >= tZ) { cb = i; rb = tZ - cum; break; }
      cum += hist[i];
    }
    if (cb < 0) { shAll = 1; shCb = NBINS - 1; shRbin = 0; }
    else        { shAll = 0; shCb = cb;        shRbin = rb; }
  }
  __syncthreads();
  const int selAll = shAll, cb = shCb;
  const float cbLow = (float)cb * WBIN;

  // ---------------- Phase C: refine inside the crossing bin ----------------
  for (int i = tid; i < SUBBINS; i += NTHREADS) hist[i] = 0ull;  // reuse LDS
  __syncthreads();
  if (!selAll) {
    for (int i = tid; i < nV4; i += NTHREADS) {
      float4 v = row4[i];
#pragma unroll
      for (int k = 0; k < 4; ++k) {
        float d = fmaxf(m - ((const float*)&v)[k] * invT, 0.0f);
        int bin = (int)(d * BINSCALE); bin = bin < (NBINS - 1) ? bin : (NBINS - 1);
        if (bin == cb) {
          int sub = (int)((d - cbLow) * SUBSCALE);
          sub = sub < 0 ? 0 : (sub > (SUBBINS - 1) ? (SUBBINS - 1) : sub);
          unsigned long long wf = (unsigned long long)(__expf(-d) * FIXSCALE);
          if (wf) atomicAdd(&hist[sub], wf);
        }
      }
    }
    for (int i = (nV4 << 2) + tid; i < V; i += NTHREADS) {
      float d = fmaxf(m - row[i] * invT, 0.0f);
      int bin = (int)(d * BINSCALE); bin = bin < (NBINS - 1) ? bin : (NBINS - 1);
      if (bin == cb) {
        int sub = (int)((d - cbLow) * SUBSCALE);
        sub = sub < 0 ? 0 : (sub > (SUBBINS - 1) ? (SUBBINS - 1) : sub);
        unsigned long long wf = (unsigned long long)(__expf(-d) * FIXSCALE);
        if (wf) atomicAdd(&hist[sub], wf);
      }
    }
  }
  __syncthreads();
  if (tid == 0) {
    int sj = SUBBINS - 1;
    if (!selAll) {
      unsigned long long cum = 0, rb = shRbin;
      for (int i = 0; i < SUBBINS; ++i) {
        cum += hist[i];
        if (cum >= rb) { sj = i; break; }
      }
    }
    shSj = sj;
  }
  __syncthreads();
  Sel S; S.m = m; S.invT = invT; S.cbLow = cbLow;
  S.cb = cb; S.sj = shSj; S.all = selAll;

  // ---------------- Phase D: contiguous-chunk selected-mass sums -----------
  int ch = (V + 255) / 256; ch = (ch + 127) & ~127; if (ch < 128) ch = 128;
  const int nch = (V + ch - 1) / ch;                 // <= 256
  for (int c = wid; c < nch; c += NWAVES) {
    float s = 0.0f;
    const int base = c * ch;
    for (int e = lane * 4; e < ch; e += 128) {
      const int idx = base + e;
      if (idx + 3 < V) {
        float4 v = *(const float4*)(row + idx);
        s += selW(v.x, S) + selW(v.y, S) + selW(v.z, S) + selW(v.w, S);
      } else {
#pragma unroll
        for (int k = 0; k < 4; ++k)
          if (idx + k < V) s += selW(row[idx + k], S);
      }
    }
    s = waveSumF(s);
    if (lane == 0) chunkSum[c] = s;
  }
  __syncthreads();
  if (tid == 0) {
    double tot = 0.0;
    for (int c = 0; c < nch; ++c) tot += (double)chunkSum[c];
    double target = (double)uvec[b] * tot;           // u * Z_sel
    double cum = 0.0; int cross = -1; float rem = 0.0f;
    for (int c = 0; c < nch; ++c) {
      if (cum + (double)chunkSum[c] >= target) {
        cross = c; rem = (float)(target - cum); break;
      }
      cum += (double)chunkSum[c];
    }
    shCross = cross; shRem = rem; shToken = V - 1;   // default: clamp
  }
  __syncthreads();

  // ---------------- Phase E: exact index inside crossing chunk (wave 0) ----
  if (wid == 0 && shCross >= 0) {
    const int c = shCross; const float rem = shRem;
    const int base = c * ch;
    bool found = false;
    float carry = 0.0f;
    for (int e = 0; e < ch && !found; e += 32) {
      const int idx = base + e + lane;
      float w = (idx < V) ? selW(row[idx], S) : 0.0f;
      float v = w;                                   // inclusive wave scan
#pragma unroll
      for (int o = 1; o < 32; o <<= 1) {
        float t = __shfl_up(v, o, 32);
        if (lane >= o) v += t;
      }
      float tot = carry + v;
      unsigned long long mask = __ballot(tot >= rem);
      if (mask) {
        int fl = __ffsll(mask) - 1;                  // token = first i with cdf>=u
        int tok = base + e + fl;
        if (lane == 0) shToken = tok < (V - 1) ? tok : (V - 1);
        found = true;
      }
      carry += __shfl(v, 31, 32);
    }
  }
  __syncthreads();
  if (tid == 0) out[b] = shToken;                    // reference output: int32
}

extern "C" void kernel_launch(void* const* d_in, const int* in_sizes, int n_in,
                              void* d_out, int out_size, void* d_ws, size_t ws_size,
                              hipStream_t stream) {
  const float* logits = (const float*)d_in[0];
  const float* u      = (const float*)d_in[1];
  const int*   temp   = (const int*)d_in[2];
  const int B = in_sizes[1];
  const int V = in_sizes[0] / B;
  int* out = (int*)d_out;
  (void)n_in; (void)out_size; (void)d_ws; (void)ws_size;
  NucleusSampling_79336635892529_kernel<<<dim3(B), dim3(NTHREADS), 0, stream>>>(
      logits, u, temp, out, V);
}